// ZBLRepulsionEnergy_2516850835917
// MI455X (gfx1250) — compile-verified
//
#include <hip/hip_runtime.h>
#include <math.h>

typedef float v2f __attribute__((ext_vector_type(2)));
typedef float v8f __attribute__((ext_vector_type(8)));
typedef int   v4i __attribute__((ext_vector_type(4)));

/* pointer to v4i in global (AS1) / LDS (AS3) address space */
typedef __attribute__((address_space(1))) v4i* as1_v4i_ptr;
typedef __attribute__((address_space(3))) v4i* as3_v4i_ptr;

#define NATOMS           8192
#define KNBR             128
#define NBATCH           16
#define ATOMS_PER_BLOCK  256
#define BLOCKS_PER_BATCH (NATOMS / ATOMS_PER_BLOCK)   /* 32 */
#define THREADS          512                          /* 16 waves of 32 */
#define NV4              (ATOMS_PER_BLOCK * KNBR / 4) /* 8192 float4 groups per block */

#define ZTAB_WS_OFFSET   4096                         /* bytes; partials live at 0 */

__device__ __forceinline__ float softplusf(float x) {
    // parameters are O(1); plain log1p(exp(x)) is safe and accurate here
    return log1pf(expf(x));
}

// ---------------------------------------------------------------------------
// Stage 0: precompute the per-atom table {Z^sp(apow), Z} once (131072 powf
// total instead of 4.2M when every stage-1 block re-derives it).
// ---------------------------------------------------------------------------
__global__ __launch_bounds__(256)
void zbl_precompute(const int* __restrict__ anum,
                    const float* __restrict__ p_apow,
                    float2* __restrict__ ztab)
{
    const int i = blockIdx.x * 256 + threadIdx.x;   // 0 .. B*N-1
    const float sp_apow = softplusf(p_apow[0]);
    const int z = anum[i];
    float2 v;
    v.x = powf((float)z, sp_apow);
    v.y = (float)z;
    ztab[i] = v;
}

// ---------------------------------------------------------------------------
// Stage 1: per-block partial sums.
// grid = (BLOCKS_PER_BATCH, NBATCH), block = 512 threads (16 waves).
// LDS: whole-batch table s_tab[n] = {Z_n^sp(apow), Z_n} (float2, 64KB) so all
// neighbor gathers are one ds_load_b64; global traffic is pure streaming b128.
// The table fill uses the CDNA5 async global->LDS DMA path (ASYNCcnt):
// no VGPR staging, one instruction per 16 bytes.
// ---------------------------------------------------------------------------
__global__ __launch_bounds__(THREADS)
void zbl_stage1(const int*    __restrict__ neighbors,
                const float*  __restrict__ nmask,
                const float2* __restrict__ ztab,
                const float*  __restrict__ dist,
                const float* __restrict__ p_adiv,
                const float* __restrict__ p_c1,   const float* __restrict__ p_c2,
                const float* __restrict__ p_c3,   const float* __restrict__ p_c4,
                const float* __restrict__ p_a1,   const float* __restrict__ p_a2,
                const float* __restrict__ p_a3,   const float* __restrict__ p_a4,
                float* __restrict__ partials)
{
    __shared__ float2 s_tab[NATOMS];                  // 64 KB
    __shared__ float  s_red[THREADS / 32];

    const int t  = threadIdx.x;
    const int bx = blockIdx.x;   // 256-atom chunk within batch
    const int b  = blockIdx.y;   // batch

    // ---- scalar parameter prep (uniform; negligible cost) ----
    const float LOG2E   = 1.4426950408889634f;
    const float KEHALF  = 7.1998226759752742f;
    const float sp_adiv = softplusf(p_adiv[0]);
    const float c1s = softplusf(p_c1[0]);
    const float c2s = softplusf(p_c2[0]);
    const float c3s = softplusf(p_c3[0]);
    const float c4s = softplusf(p_c4[0]);
    const float inv_csum = 1.0f / (c1s + c2s + c3s + c4s);
    const float w1 = c1s * inv_csum, w2 = c2s * inv_csum;
    const float w3 = c3s * inv_csum, w4 = c4s * inv_csum;
    // exp(-a_k * x) == v_exp_f32(e_k * x)
    const float e1 = -softplusf(p_a1[0]) * LOG2E;
    const float e2 = -softplusf(p_a2[0]) * LOG2E;
    const float e3 = -softplusf(p_a3[0]) * LOG2E;
    const float e4 = -softplusf(p_a4[0]) * LOG2E;

    // ---- stage whole-batch {Z^p, Z} table into LDS ----
    {
        const float4* __restrict__ zrow4 =
            (const float4*)(ztab + (size_t)b * NATOMS);        // 4096 x 16B
        float4* s_tab4 = (float4*)s_tab;
#if __has_builtin(__builtin_amdgcn_global_load_async_to_lds_b128)
        for (int i = t; i < NATOMS / 2; i += THREADS) {
            __builtin_amdgcn_global_load_async_to_lds_b128(
                (as1_v4i_ptr)(zrow4 + i),
                (as3_v4i_ptr)(s_tab4 + i),
                /*offset=*/0, /*cpol=*/0);
        }
#if __has_builtin(__builtin_amdgcn_s_wait_asynccnt)
        __builtin_amdgcn_s_wait_asynccnt(0);
#else
        asm volatile("s_wait_asynccnt 0" ::: "memory");
#endif
#else
        for (int i = t; i < NATOMS / 2; i += THREADS)
            s_tab4[i] = zrow4[i];
#endif
    }
    __syncthreads();

    // ---- stream edges: 32768 edges per block as 8192 x (int4,float4,float4) ----
    const size_t ebase = ((size_t)b * NATOMS + (size_t)bx * ATOMS_PER_BLOCK) * KNBR;
    const int4*   __restrict__ nb4 = (const int4*)(neighbors + ebase);
    const float4* __restrict__ mk4 = (const float4*)(nmask + ebase);
    const float4* __restrict__ ds4 = (const float4*)(dist + ebase);

    float acc = 0.0f;

#define ZBL_EDGE(J, M, D)                                                     \
    do {                                                                      \
        float2 zj = s_tab[(J)];                                               \
        float a   = (zpi + zj.x) * sp_adiv * (M);                             \
        float zz  = zfi * zj.y * (M);                                         \
        float x   = a * (D);                                                  \
        float f   = w1 * __builtin_amdgcn_exp2f(e1 * x)                       \
                  + w2 * __builtin_amdgcn_exp2f(e2 * x)                       \
                  + w3 * __builtin_amdgcn_exp2f(e3 * x)                       \
                  + w4 * __builtin_amdgcn_exp2f(e4 * x);                      \
        float c   = (KEHALF * f * zz) * __builtin_amdgcn_rcpf((D));           \
        acc += ((M) != 0.0f) ? c : 0.0f;                                      \
    } while (0)

    #pragma unroll 1
    for (int idx = t; idx < NV4; idx += THREADS) {
        // guarded in-bounds prefetch of the next iteration's cache lines
        if (idx + THREADS < NV4) {
            __builtin_prefetch((const void*)(nb4 + idx + THREADS), 0, 1);
            __builtin_prefetch((const void*)(mk4 + idx + THREADS), 0, 1);
            __builtin_prefetch((const void*)(ds4 + idx + THREADS), 0, 1);
        }
        int4   j4 = nb4[idx];
        float4 m4 = mk4[idx];
        float4 d4 = ds4[idx];

        int    row = idx >> 5;                      // (idx*4)/128, wave-uniform
        float2 zi  = s_tab[bx * ATOMS_PER_BLOCK + row];
        float  zpi = zi.x;
        float  zfi = zi.y;

        ZBL_EDGE(j4.x, m4.x, d4.x);
        ZBL_EDGE(j4.y, m4.y, d4.y);
        ZBL_EDGE(j4.z, m4.z, d4.z);
        ZBL_EDGE(j4.w, m4.w, d4.w);
    }
#undef ZBL_EDGE

    // ---- wave32 shuffle reduce, then cross-wave LDS reduce ----
    for (int off = 16; off > 0; off >>= 1)
        acc += __shfl_down(acc, off, 32);
    if ((t & 31) == 0) s_red[t >> 5] = acc;
    __syncthreads();
    if (t == 0) {
        float s = 0.0f;
        #pragma unroll
        for (int w = 0; w < THREADS / 32; ++w) s += s_red[w];
        partials[b * BLOCKS_PER_BATCH + bx] = s;
    }
}

// ---------------------------------------------------------------------------
// Stage 2: deterministic exact f32 reduction of the 16x32 partial matrix via
// 8 chained V_WMMA_F32_16X16X4_F32 against an all-ones B:
//   D[m][n] = sum_k A[m][k] * 1 + C  ->  after 8 steps D[b][*] = sum of 32.
// One full wave32, EXEC all ones (WMMA requirement).
// ---------------------------------------------------------------------------
__global__ __launch_bounds__(32)
void zbl_stage2(const float* __restrict__ part, float* __restrict__ out)
{
    const int lane = threadIdx.x;
    const int m    = lane & 15;
    const int kofs = (lane < 16) ? 0 : 2;   // A 16x4 f32 layout: lanes16-31 carry K=2,3

    v8f acc = {};
    v2f ones;
    ones.x = 1.0f; ones.y = 1.0f;

    #pragma unroll
    for (int c = 0; c < 8; ++c) {
        v2f a;
        a.x = part[m * BLOCKS_PER_BATCH + 4 * c + kofs];
        a.y = part[m * BLOCKS_PER_BATCH + 4 * c + kofs + 1];
        acc = __builtin_amdgcn_wmma_f32_16x16x4_f32(
            /*neg_a=*/false, a, /*neg_b=*/false, ones,
            /*c_mod=*/(short)0, acc, /*reuse_a=*/false, /*reuse_b=*/false);
    }

    // Column N=0 of D: lane 0 holds M=0..7 in regs 0..7, lane 16 holds M=8..15.
    if (lane == 0) {
        #pragma unroll
        for (int i = 0; i < 8; ++i) out[i] = acc[i];
    } else if (lane == 16) {
        #pragma unroll
        for (int i = 0; i < 8; ++i) out[8 + i] = acc[i];
    }
}

// ---------------------------------------------------------------------------
extern "C" void kernel_launch(void* const* d_in, const int* in_sizes, int n_in,
                              void* d_out, int out_size, void* d_ws, size_t ws_size,
                              hipStream_t stream)
{
    const int*   neighbors = (const int*)d_in[0];
    const float* nmask     = (const float*)d_in[1];
    const int*   anum      = (const int*)d_in[2];
    const float* dist      = (const float*)d_in[3];
    const float* p_adiv    = (const float*)d_in[4];
    const float* p_apow    = (const float*)d_in[5];
    const float* p_c1      = (const float*)d_in[6];
    const float* p_c2      = (const float*)d_in[7];
    const float* p_c3      = (const float*)d_in[8];
    const float* p_c4      = (const float*)d_in[9];
    const float* p_a1      = (const float*)d_in[10];
    const float* p_a2      = (const float*)d_in[11];
    const float* p_a3      = (const float*)d_in[12];
    const float* p_a4      = (const float*)d_in[13];

    float*  partials = (float*)d_ws;                               // 512 floats
    float2* ztab     = (float2*)((char*)d_ws + ZTAB_WS_OFFSET);    // B*N float2 = 1 MB
    float*  out      = (float*)d_out;                              // [B,1] -> 16 floats

    zbl_precompute<<<(NBATCH * NATOMS) / 256, 256, 0, stream>>>(anum, p_apow, ztab);

    dim3 grid1(BLOCKS_PER_BATCH, NBATCH);    // (32, 16)
    zbl_stage1<<<grid1, THREADS, 0, stream>>>(
        neighbors, nmask, ztab, dist,
        p_adiv, p_c1, p_c2, p_c3, p_c4, p_a1, p_a2, p_a3, p_a4,
        partials);

    zbl_stage2<<<1, 32, 0, stream>>>(partials, out);

    (void)in_sizes; (void)n_in; (void)out_size; (void)ws_size;
}